// Decoder_58557584114356
// MI455X (gfx1250) — compile-verified
//
#include <hip/hip_runtime.h>
#include <math.h>

typedef __attribute__((ext_vector_type(2))) float v2f;
typedef __attribute__((ext_vector_type(8))) float v8f;

#define BB 32
#define NN 100000
#define LL 12
#define SS 256

// ---------------------------------------------------------------------------
// Pose precompute: R rows 0,1 (ZYZ) and shifts (+S/2 folded in). pose[b*8+..]
// ---------------------------------------------------------------------------
__global__ void prep_pose(const float* __restrict__ euler_base,
                          const float* __restrict__ delta_euler,
                          const float* __restrict__ shifts_base,
                          const float* __restrict__ delta_shifts,
                          float* __restrict__ pose) {
  int b = threadIdx.x;
  if (b >= BB) return;
  float rot  = euler_base[b*3+0] + delta_euler[b*3+0];
  float tilt = euler_base[b*3+1] + delta_euler[b*3+1];
  float psi  = euler_base[b*3+2] + delta_euler[b*3+2];
  float ca = cosf(rot),  sa = sinf(rot);
  float cb = cosf(tilt), sb = sinf(tilt);
  float cg = cosf(psi),  sg = sinf(psi);
  pose[b*8+0] =  cg*cb*ca - sg*sa;
  pose[b*8+1] =  cg*cb*sa + sg*ca;
  pose[b*8+2] = -cg*sb;
  pose[b*8+3] = -sg*cb*ca - cg*sa;
  pose[b*8+4] = -sg*cb*sa + cg*ca;
  pose[b*8+5] =  sg*sb;
  pose[b*8+6] = shifts_base[b*2+0] + delta_shifts[b*2+0] + 128.0f;
  pose[b*8+7] = shifts_base[b*2+1] + delta_shifts[b*2+1] + 128.0f;
}

// ---------------------------------------------------------------------------
// DFT matrices: Wr[j,k]=cos(2*pi*j*k/256), Wi[j,k]=-sin(2*pi*j*k/256)
// (symmetric; exact periodic index reduction)
// ---------------------------------------------------------------------------
__global__ void build_W(float* __restrict__ Wr, float* __restrict__ Wi) {
  int j = blockIdx.x, k = threadIdx.x;
  int m = (j * k) & 255;
  float th = (float)m * 0.024543692606170259f;  // 2*pi/256
  Wr[j*256 + k] = cosf(th);
  Wi[j*256 + k] = -sinf(th);
}

__global__ void zero_f32(float* __restrict__ p, int n) {
  int i = blockIdx.x * blockDim.x + threadIdx.x;
  if (i < n) p[i] = 0.0f;
}

// ---------------------------------------------------------------------------
// Deform + bonds + angles + bilinear splat.
// One thread per atom; loop over batches; per-batch p tile (+halo) in LDS.
// ---------------------------------------------------------------------------
__global__ __launch_bounds__(128)
void deform_kernel(const float* __restrict__ c_nma, const float* __restrict__ coords,
                   const float* __restrict__ bx, const float* __restrict__ by,
                   const float* __restrict__ bz, const float* __restrict__ pose,
                   float* __restrict__ img, float* __restrict__ out_bond,
                   float* __restrict__ out_angle) {
  __shared__ float sh_c[BB*LL];
  __shared__ float sh_pose[BB*8];
  __shared__ float sh_p[130*3];

  int t = threadIdx.x;
  int tile0 = blockIdx.x * 128;
  int a = tile0 + t;

  for (int i = t; i < BB*LL; i += 128) sh_c[i] = c_nma[i];
  for (int i = t; i < BB*8;  i += 128) sh_pose[i] = pose[i];

  bool valid = (a < NN);
  float rbx[LL], rby[LL], rbz[LL];
  float cx = 0.f, cy = 0.f, cz = 0.f;
  if (valid) {
    cx = coords[a*3+0]; cy = coords[a*3+1]; cz = coords[a*3+2];
    #pragma unroll
    for (int l = 0; l < LL; ++l) {
      rbx[l] = bx[a*LL+l]; rby[l] = by[a*LL+l]; rbz[l] = bz[a*LL+l];
    }
  }

  // halo atoms: thread 0 -> tile0-1 (LDS slot 0), thread 1 -> tile0+128 (slot 129)
  int ha = -1;
  if (t == 0 && tile0 > 0)        ha = tile0 - 1;
  if (t == 1 && tile0 + 128 < NN) ha = tile0 + 128;
  int hslot = (t == 0) ? 0 : 129;
  float hbx[LL], hby[LL], hbz[LL];
  float hcx = 0.f, hcy = 0.f, hcz = 0.f;
  if (ha >= 0) {
    hcx = coords[ha*3+0]; hcy = coords[ha*3+1]; hcz = coords[ha*3+2];
    #pragma unroll
    for (int l = 0; l < LL; ++l) {
      hbx[l] = bx[ha*LL+l]; hby[l] = by[ha*LL+l]; hbz[l] = bz[ha*LL+l];
    }
  }
  __syncthreads();

  for (int b = 0; b < BB; ++b) {
    const float* cb = &sh_c[b*LL];
    float px = cx, py = cy, pz = cz;
    if (valid) {
      #pragma unroll
      for (int l = 0; l < LL; ++l) {
        px += rbx[l]*cb[l]; py += rby[l]*cb[l]; pz += rbz[l]*cb[l];
      }
      sh_p[(t+1)*3+0] = px; sh_p[(t+1)*3+1] = py; sh_p[(t+1)*3+2] = pz;
    }
    if (ha >= 0) {
      float qx = hcx, qy = hcy, qz = hcz;
      #pragma unroll
      for (int l = 0; l < LL; ++l) {
        qx += hbx[l]*cb[l]; qy += hby[l]*cb[l]; qz += hbz[l]*cb[l];
      }
      sh_p[hslot*3+0] = qx; sh_p[hslot*3+1] = qy; sh_p[hslot*3+2] = qz;
    }
    __syncthreads();

    if (a < NN-1) {  // bond between a and a+1
      float dx = sh_p[(t+2)*3+0] - px;
      float dy = sh_p[(t+2)*3+1] - py;
      float dz = sh_p[(t+2)*3+2] - pz;
      out_bond[(size_t)b*(NN-1) + a] = sqrtf(dx*dx + dy*dy + dz*dz + 1e-12f);
    }
    if (a >= 1 && a <= NN-2) {  // angle at interior atom a
      float ux = sh_p[t*3+0] - px, uy = sh_p[t*3+1] - py, uz = sh_p[t*3+2] - pz;
      float vx = sh_p[(t+2)*3+0] - px, vy = sh_p[(t+2)*3+1] - py, vz = sh_p[(t+2)*3+2] - pz;
      float uu = ux*ux + uy*uy + uz*uz;
      float vv = vx*vx + vy*vy + vz*vz;
      float cosang = (ux*vx + uy*vy + uz*vz) / (sqrtf(uu)*sqrtf(vv) + 1e-12f);
      cosang = fminf(fmaxf(cosang, -0.9999f), 0.9999f);
      out_angle[(size_t)b*(NN-2) + (a-1)] = acosf(cosang);
    }
    if (valid) {
      const float* P = &sh_pose[b*8];
      float x = P[0]*px + P[1]*py + P[2]*pz + P[6];
      float y = P[3]*px + P[4]*py + P[5]*pz + P[7];
      x = fminf(fmaxf(x, 0.0f), 254.999f);
      y = fminf(fmaxf(y, 0.0f), 254.999f);
      int x0 = (int)x, y0 = (int)y;
      float fx = x - (float)x0, fy = y - (float)y0;
      float* ib = img + (size_t)b*65536;
      atomicAdd(&ib[y0*256 + x0],         (1.0f-fx)*(1.0f-fy));
      atomicAdd(&ib[y0*256 + x0 + 1],     fx*(1.0f-fy));
      atomicAdd(&ib[(y0+1)*256 + x0],     (1.0f-fx)*fy);
      atomicAdd(&ib[(y0+1)*256 + x0 + 1], fx*fy);
    }
    __syncthreads();
  }
}

// ---------------------------------------------------------------------------
// WMMA stage: D_storeT = s1*A1*B1 (+ s2*A2*B2), optional on-the-fly
// Gaussian*CTF*(1/S^2) filter in the epilogue. B1/B2 are symmetric 256x256
// (Wr/Wi) so fragments load as contiguous float2. A/D have per-batch stride
// 65536. One wave per 16x16 output tile, K-loop of 64 x V_WMMA_F32_16X16X4_F32.
// ---------------------------------------------------------------------------
template <bool HASA2, bool USEH>
__global__ __launch_bounds__(128)
void wmma_stage(const float* __restrict__ A1, const float* __restrict__ B1, float s1,
                const float* __restrict__ A2, const float* __restrict__ B2, float s2,
                float* __restrict__ D, const float* __restrict__ defocus) {
  int gw   = blockIdx.x * 4 + (threadIdx.x >> 5);
  int lane = threadIdx.x & 31;
  int b    = gw >> 8;
  int tile = gw & 255;
  int tym = tile >> 4, txn = tile & 15;
  int halfSel = lane >> 4;
  int lm = lane & 15;
  int mRow = tym*16 + lm;   // A row this lane supplies
  int nCol = txn*16 + lm;   // B col this lane supplies

  const float* A1b = A1 + (size_t)b * 65536;
  const float* A2b = HASA2 ? (A2 + (size_t)b * 65536) : A1;

  v8f acc  = {};
  v8f acc2 = {};
  for (int k0 = 0; k0 < 256; k0 += 4) {
    int ka = k0 + halfSel * 2;
    v2f av = *(const v2f*)(A1b + mRow*256 + ka);
    v2f bv = *(const v2f*)(B1  + nCol*256 + ka);   // B symmetric: B[k][n]==B[n][k]
    acc = __builtin_amdgcn_wmma_f32_16x16x4_f32(false, av, false, bv,
                                                (short)0, acc, false, false);
    if (HASA2) {
      v2f av2 = *(const v2f*)(A2b + mRow*256 + ka);
      v2f bv2 = *(const v2f*)(B2  + nCol*256 + ka);
      acc2 = __builtin_amdgcn_wmma_f32_16x16x4_f32(false, av2, false, bv2,
                                                   (short)0, acc2, false, false);
    }
  }

  float* Db = D + (size_t)b * 65536;
  float fn2 = 0.f, defb = 0.f;
  if (USEH) {
    float fn = (float)((nCol < 128) ? nCol : nCol - 256) * (1.0f/256.0f);
    fn2 = fn * fn;
    defb = defocus[b];
  }
  const float GCOEF = (float)(-2.0 * M_PI * M_PI);               // sigma=1
  const float C1    = (float)(M_PI * 0.0197);                     // pi*lambda
  const float C2    = (float)(0.5 * M_PI * 2.7e7 * 0.0197 * 0.0197 * 0.0197);
  const float SQ    = 0.994987437106620f;                         // sqrt(1-Q0^2)

  #pragma unroll
  for (int r = 0; r < 8; ++r) {
    int pM = tym*16 + halfSel*8 + r;
    float v = s1 * acc[r];
    if (HASA2) v += s2 * acc2[r];
    if (USEH) {
      float fm = (float)((pM < 128) ? pM : pM - 256) * (1.0f/256.0f);
      float ss = fn2 + fm*fm;
      float g = expf(GCOEF * ss);
      float chi = C1 * defb * ss - C2 * ss * ss;
      float ctf = -(SQ * sinf(chi) + 0.1f * cosf(chi));
      v *= g * ctf * (1.0f/65536.0f);
    }
    Db[(size_t)nCol*256 + pM] = v;   // store transposed
  }
}

// ---------------------------------------------------------------------------
extern "C" void kernel_launch(void* const* d_in, const int* in_sizes, int n_in,
                              void* d_out, int out_size, void* d_ws, size_t ws_size,
                              hipStream_t stream) {
  const float* c_nma        = (const float*)d_in[0];
  const float* delta_euler  = (const float*)d_in[1];
  const float* delta_shifts = (const float*)d_in[2];
  const float* coords       = (const float*)d_in[3];
  const float* basis_x      = (const float*)d_in[4];
  const float* basis_y      = (const float*)d_in[5];
  const float* basis_z      = (const float*)d_in[6];
  const float* euler_base   = (const float*)d_in[7];
  const float* shifts_base  = (const float*)d_in[8];
  const float* defocus      = (const float*)d_in[9];

  float* out       = (float*)d_out;
  float* out_dec   = out;                                   // 32*256*256
  float* out_bond  = out + (size_t)BB * 65536;              // 32*99999
  float* out_angle = out_bond + (size_t)BB * (NN - 1);      // 32*99998

  float* ws   = (float*)d_ws;
  float* Wr   = ws;
  float* Wi   = Wr + 65536;
  float* pose = Wi + 65536;                 // 256 floats
  float* X    = ws + 131328;                // slot0: img  [32][65536]
  float* G1   = X  + 2097152;               // slot1
  float* G2   = G1 + 2097152;               // slot2
  float* Zi   = G2 + 2097152;               // slot3
  float* Zr   = X;                          // reuse slot0 (img dead after S1)
  float* Mr   = G1;                         // reuse slot1 (G dead after S2)
  float* Mi   = G2;                         // reuse slot2

  build_W<<<256, 256, 0, stream>>>(Wr, Wi);
  prep_pose<<<1, 32, 0, stream>>>(euler_base, delta_euler, shifts_base,
                                  delta_shifts, pose);
  zero_f32<<<(2097152 + 255) / 256, 256, 0, stream>>>(X, 2097152);
  deform_kernel<<<(NN + 127) / 128, 128, 0, stream>>>(
      c_nma, coords, basis_x, basis_y, basis_z, pose, X, out_bond, out_angle);

  // Forward 2D DFT: A = X*W (store transposed)
  wmma_stage<false, false><<<2048, 128, 0, stream>>>(X, Wr, 1.f, nullptr, nullptr, 0.f, G1, nullptr);
  wmma_stage<false, false><<<2048, 128, 0, stream>>>(X, Wi, 1.f, nullptr, nullptr, 0.f, G2, nullptr);
  // Y = W*A, then Z = H .* Y / S^2 (filter fused in epilogue)
  wmma_stage<true, true ><<<2048, 128, 0, stream>>>(G1, Wr, 1.f, G2, Wi, -1.f, Zr, defocus);
  wmma_stage<true, true ><<<2048, 128, 0, stream>>>(G1, Wi, 1.f, G2, Wr,  1.f, Zi, defocus);
  // Inverse: M = Z*V (V = conj(W)), stored transposed
  wmma_stage<true, false><<<2048, 128, 0, stream>>>(Zr, Wr, 1.f, Zi, Wi,  1.f, Mr, nullptr);
  wmma_stage<true, false><<<2048, 128, 0, stream>>>(Zi, Wr, 1.f, Zr, Wi, -1.f, Mi, nullptr);
  // out = Re(V*M)
  wmma_stage<true, false><<<2048, 128, 0, stream>>>(Mr, Wr, 1.f, Mi, Wi,  1.f, out_dec, nullptr);
}